// GAT_35321811043078
// MI455X (gfx1250) — compile-verified
//
#include <hip/hip_runtime.h>
#include <hip/hip_bf16.h>
#include <math.h>

typedef __attribute__((ext_vector_type(2))) float v2f;
typedef __attribute__((ext_vector_type(8))) float v8f;
typedef __attribute__((ext_vector_type(4))) unsigned int u32x4;
typedef __attribute__((ext_vector_type(8))) unsigned int u32x8;
typedef __attribute__((address_space(3))) void lds_void;

#define NEG_SLOPE 0.2f
#define GN_EPS 1e-5f

// ---------------------------------------------------------------------------
// GEMM: Out[m, d] = sum_k X[m,k] * W[d,k]   (C = X * W^T)
// X: [nRows, K] row-major, W: [D, K] row-major, Out: [nRows, D]
// Block = 256 threads (8 waves). The block's 64-column W tile (64 x K floats,
// <= 64KB) is staged into LDS by ONE tensor_load_to_lds (TDM) issued by wave 0,
// synchronized with s_wait_tensorcnt. Each wave then computes a 16x64 strip
// with V_WMMA_F32_16X16X4_F32, reading B fragments from LDS (ds_load_b64)
// and software-pipelining the global A-fragment loads.
// Fragment layouts (wave32):
//   A (16x4 f32): lanes 0-15 -> M=0..15, VGPR j holds K = 2*half + j.
//   B (4x16 f32): lane col N = lane%16, VGPR j holds K = 2*half + j.
//   C/D (16x16 f32): VGPR v -> row M = v + 8*half, col N = lane%16.
// ---------------------------------------------------------------------------
__global__ __launch_bounds__(256)
void gat_gemm_wmma(const float* __restrict__ X, const float* __restrict__ W,
                   float* __restrict__ Out, int K, int D, int nRows) {
  __shared__ float smemW[64 * 256];        // 64 cols x K floats (K <= 256)

  const int n0   = blockIdx.y * 64;
  const int wave = threadIdx.x >> 5;
  const int lane = threadIdx.x & 31;
  const int hf   = lane >> 4;              // 0 or 1
  const int r    = lane & 15;

  // ---- TDM stage: W[n0 .. n0+63][0 .. K) -> smemW (wave 0 only) ----
  if (wave == 0) {
    const unsigned long long ga =
        (unsigned long long)(const void*)(W + (size_t)n0 * K);
    const unsigned ldsAddr =
        (unsigned)(unsigned long long)(lds_void*)&smemW[0];
    const unsigned uK = (unsigned)K;
    const unsigned uD = (unsigned)D;

    u32x4 g0;
    g0[0] = 1u;                                         // count=1, user mode
    g0[1] = ldsAddr;                                    // lds_addr (bytes)
    g0[2] = (unsigned)(ga & 0xFFFFFFFFu);               // global_addr[31:0]
    g0[3] = (unsigned)((ga >> 32) & 0x1FFFFFFu)         // global_addr[56:32]
          | (2u << 30);                                 // type = 2 (image)

    u32x8 g1;
    g1[0] = (2u << 16);                                 // data_size = 4 bytes
    g1[1] = (uK & 0xFFFFu) << 16;                       // tensor_dim0[15:0]
    g1[2] = ((uK >> 16) & 0xFFFFu) | ((uD & 0xFFFFu) << 16);  // dim0 hi, dim1 lo
    g1[3] = ((uD >> 16) & 0xFFFFu) | ((uK & 0xFFFFu) << 16);  // dim1 hi, tile_dim0=K
    g1[4] = 64u;                                        // tile_dim1=64, tile_dim2=0
    g1[5] = uK;                                         // dim0_stride[31:0] = K
    g1[6] = 0u;                                         // dim0_stride hi, dim1_stride lo
    g1[7] = 0u;                                         // dim1_stride hi (unused, 2D)

    asm volatile("tensor_load_to_lds %0, %1" :: "s"(g0), "s"(g1) : "memory");
    __builtin_amdgcn_s_wait_tensorcnt(0);
  }
  __syncthreads();

  const int m0 = (blockIdx.x * 8 + wave) * 16;
  if (m0 < nRows) {
    const float* xrow = X + (size_t)(m0 + r) * K;       // A rows (global)
    const float* wc   = &smemW[(size_t)r * K];          // B col base (LDS)

    v8f acc0 = {}, acc1 = {}, acc2 = {}, acc3 = {};

    v2f a = *(const v2f*)(xrow + 2 * hf);               // prefetch first A frag
    int kk = 0;
    for (; kk + 4 < K; kk += 4) {
      v2f an = *(const v2f*)(xrow + (kk + 4) + 2 * hf); // prefetch next A frag
      const int ko = kk + 2 * hf;
      v2f b0 = *(const v2f*)(wc + (size_t)0 * 16 * K + ko);
      v2f b1 = *(const v2f*)(wc + (size_t)1 * 16 * K + ko);
      v2f b2 = *(const v2f*)(wc + (size_t)2 * 16 * K + ko);
      v2f b3 = *(const v2f*)(wc + (size_t)3 * 16 * K + ko);
      acc0 = __builtin_amdgcn_wmma_f32_16x16x4_f32(false, a, false, b0, (short)0, acc0, false, false);
      acc1 = __builtin_amdgcn_wmma_f32_16x16x4_f32(false, a, false, b1, (short)0, acc1, false, false);
      acc2 = __builtin_amdgcn_wmma_f32_16x16x4_f32(false, a, false, b2, (short)0, acc2, false, false);
      acc3 = __builtin_amdgcn_wmma_f32_16x16x4_f32(false, a, false, b3, (short)0, acc3, false, false);
      a = an;
    }
    {
      const int ko = kk + 2 * hf;
      v2f b0 = *(const v2f*)(wc + (size_t)0 * 16 * K + ko);
      v2f b1 = *(const v2f*)(wc + (size_t)1 * 16 * K + ko);
      v2f b2 = *(const v2f*)(wc + (size_t)2 * 16 * K + ko);
      v2f b3 = *(const v2f*)(wc + (size_t)3 * 16 * K + ko);
      acc0 = __builtin_amdgcn_wmma_f32_16x16x4_f32(false, a, false, b0, (short)0, acc0, false, false);
      acc1 = __builtin_amdgcn_wmma_f32_16x16x4_f32(false, a, false, b1, (short)0, acc1, false, false);
      acc2 = __builtin_amdgcn_wmma_f32_16x16x4_f32(false, a, false, b2, (short)0, acc2, false, false);
      acc3 = __builtin_amdgcn_wmma_f32_16x16x4_f32(false, a, false, b3, (short)0, acc3, false, false);
    }

#pragma unroll
    for (int v = 0; v < 8; ++v) {
      const int row = m0 + v + 8 * hf;
      float* orow = Out + (size_t)row * D + n0 + r;
      orow[0]  = acc0[v];
      orow[16] = acc1[v];
      orow[32] = acc2[v];
      orow[48] = acc3[v];
    }
  }
}

// ---------------------------------------------------------------------------
// Attention scores: e_src[n,h] = sum_c h[n,h,c]*a_src[h,c]  (same for dst)
// ---------------------------------------------------------------------------
__global__ __launch_bounds__(256)
void gat_att_scores(const float* __restrict__ h, const float* __restrict__ a_src,
                    const float* __restrict__ a_dst, float* __restrict__ esrc,
                    float* __restrict__ edst, int n, int H, int C) {
  const int idx = blockIdx.x * blockDim.x + threadIdx.x;
  if (idx >= n * H) return;
  const int node = idx / H;
  const int hh   = idx % H;
  const float* hp = h + (size_t)node * H * C + (size_t)hh * C;
  const float* as = a_src + (size_t)hh * C;
  const float* ad = a_dst + (size_t)hh * C;
  float s = 0.f, d = 0.f;
  for (int c = 0; c < C; ++c) {
    const float v = hp[c];
    s += v * as[c];
    d += v * ad[c];
  }
  esrc[idx] = s;
  edst[idx] = d;
}

__global__ __launch_bounds__(256)
void gat_fill(float* __restrict__ p, float v, int n) {
  const int i = blockIdx.x * blockDim.x + threadIdx.x;
  if (i < n) p[i] = v;
}

// monotonic float atomic max via int/uint trick (init to -inf)
__device__ __forceinline__ void atomicMaxF(float* addr, float val) {
  if (val >= 0.0f) atomicMax((int*)addr, __float_as_int(val));
  else             atomicMin((unsigned int*)addr, __float_as_uint(val));
}

__device__ __forceinline__ void edgeSrcDst(const int* __restrict__ ei, int E,
                                           int idx, int* s, int* d) {
  if (idx < E) { *s = ei[idx]; *d = ei[E + idx]; }
  else         { *s = idx - E; *d = idx - E; }     // self-loop
}

// pass 1: segment max of leaky_relu(e_src[src]+e_dst[dst]) per (dst, head)
__global__ __launch_bounds__(256)
void gat_edge_max(const int* __restrict__ ei, int E, int n,
                  const float* __restrict__ esrc, const float* __restrict__ edst,
                  float* __restrict__ mmax, int H) {
  const int idx = blockIdx.x * blockDim.x + threadIdx.x;
  if (idx >= E + n) return;
  int s, d;
  edgeSrcDst(ei, E, idx, &s, &d);
  for (int h = 0; h < H; ++h) {
    float e = esrc[s * H + h] + edst[d * H + h];
    e = (e > 0.f) ? e : NEG_SLOPE * e;
    atomicMaxF(&mmax[d * H + h], e);
  }
}

// pass 2: ex = exp(e - max); cache per-edge ex; denom[dst,h] += ex
__global__ __launch_bounds__(256)
void gat_edge_expsum(const int* __restrict__ ei, int E, int n,
                     const float* __restrict__ esrc, const float* __restrict__ edst,
                     const float* __restrict__ mmax, float* __restrict__ exbuf,
                     float* __restrict__ denom, int H) {
  const int idx = blockIdx.x * blockDim.x + threadIdx.x;
  if (idx >= E + n) return;
  int s, d;
  edgeSrcDst(ei, E, idx, &s, &d);
  for (int h = 0; h < H; ++h) {
    float e = esrc[s * H + h] + edst[d * H + h];
    e = (e > 0.f) ? e : NEG_SLOPE * e;
    const float ex = expf(e - mmax[d * H + h]);
    exbuf[(size_t)idx * H + h] = ex;
    atomicAdd(&denom[d * H + h], ex);
  }
}

// pass 3: agg[dst, h, c] += (ex / (denom+1e-16)) * hfeat[src, h, c]
// one block per edge; blockDim = H*C
__global__ __launch_bounds__(256)
void gat_edge_agg(const int* __restrict__ ei, int E, int n,
                  const float* __restrict__ hfeat, const float* __restrict__ exbuf,
                  const float* __restrict__ denom, float* __restrict__ agg,
                  int H, int C) {
  const int idx = blockIdx.x;          // edge id
  const int tid = threadIdx.x;         // 0 .. H*C-1
  int s, d;
  edgeSrcDst(ei, E, idx, &s, &d);
  __shared__ float coef[8];
  if (tid < H)
    coef[tid] = exbuf[(size_t)idx * H + tid] / (denom[d * H + tid] + 1e-16f);
  __syncthreads();
  const int hh = tid / C;
  const float val = coef[hh] * hfeat[(size_t)s * H * C + tid];
  atomicAdd(&agg[(size_t)d * H * C + tid], val);
}

// GraphNorm column accumulation. mode 0: sum(x). mode 1: sum((x+shift)^2).
// blockDim == D (256); each block handles a row chunk; coalesced column reads.
__global__ __launch_bounds__(256)
void gat_col_accum(const float* __restrict__ x, float* __restrict__ out,
                   int n, int D, int rowsPer, int mode,
                   const float* __restrict__ shift) {
  const int d  = threadIdx.x;
  const int r0 = blockIdx.x * rowsPer;
  int r1 = r0 + rowsPer; if (r1 > n) r1 = n;
  float s = 0.f;
  if (mode == 0) {
    for (int r = r0; r < r1; ++r) s += x[(size_t)r * D + d];
  } else {
    const float sh = shift[d];
    for (int r = r0; r < r1; ++r) {
      const float v = x[(size_t)r * D + d] + sh;
      s += v * v;
    }
  }
  atomicAdd(&out[d], s);
}

// shift[d] = bias[d] - (colsum[d]/n + bias[d]) * ms[d]
__global__ __launch_bounds__(256)
void gat_prep_shift(const float* __restrict__ colsum, const float* __restrict__ bias,
                    const float* __restrict__ ms, float* __restrict__ shift, int n) {
  const int d = threadIdx.x;
  const float mean = colsum[d] / (float)n + bias[d];
  shift[d] = bias[d] - mean * ms[d];
}

// rstd[d] = rsqrt(varsum[d]/n + eps)
__global__ __launch_bounds__(256)
void gat_prep_rstd(const float* __restrict__ varsum, float* __restrict__ rstd, int n) {
  const int d = threadIdx.x;
  rstd[d] = rsqrtf(varsum[d] / (float)n + GN_EPS);
}

// x = relu(w * (agg + shift) * rstd + b), in place
__global__ __launch_bounds__(256)
void gat_apply_norm(float* __restrict__ agg, const float* __restrict__ shift,
                    const float* __restrict__ rstd, const float* __restrict__ w,
                    const float* __restrict__ b, int total, int D) {
  const int i = blockIdx.x * blockDim.x + threadIdx.x;
  if (i >= total) return;
  const int d = i % D;
  const float xc = agg[i] + shift[d];
  const float y  = w[d] * xc * rstd[d] + b[d];
  agg[i] = (y > 0.f) ? y : 0.f;
}

// y[n, o] = sum_d (agg2[n,d] + b2[d]) * ow[o,d] + ob[o]   (D=128, O=2)
__global__ __launch_bounds__(256)
void gat_final_out(const float* __restrict__ agg2, const float* __restrict__ b2,
                   const float* __restrict__ ow, const float* __restrict__ ob,
                   float* __restrict__ y, int n) {
  const int node = blockIdx.x * blockDim.x + threadIdx.x;
  if (node >= n) return;
  float a0 = 0.f, a1 = 0.f;
  const float* row = agg2 + (size_t)node * 128;
  for (int d = 0; d < 128; ++d) {
    const float v = row[d] + b2[d];
    a0 += v * ow[d];
    a1 += v * ow[128 + d];
  }
  y[node * 2 + 0] = a0 + ob[0];
  y[node * 2 + 1] = a1 + ob[1];
}

// ---------------------------------------------------------------------------
extern "C" void kernel_launch(void* const* d_in, const int* in_sizes, int n_in,
                              void* d_out, int out_size, void* d_ws, size_t ws_size,
                              hipStream_t stream) {
  const float* x   = (const float*)d_in[0];
  const int*   ei  = (const int*)d_in[1];
  const float* W0  = (const float*)d_in[2];
  const float* as0 = (const float*)d_in[3];
  const float* ad0 = (const float*)d_in[4];
  const float* b0  = (const float*)d_in[5];
  const float* gw0 = (const float*)d_in[6];
  const float* gb0 = (const float*)d_in[7];
  const float* gm0 = (const float*)d_in[8];
  const float* W1  = (const float*)d_in[9];
  const float* as1 = (const float*)d_in[10];
  const float* ad1 = (const float*)d_in[11];
  const float* b1  = (const float*)d_in[12];
  const float* gw1 = (const float*)d_in[13];
  const float* gb1 = (const float*)d_in[14];
  const float* gm1 = (const float*)d_in[15];
  const float* W2  = (const float*)d_in[16];
  const float* as2 = (const float*)d_in[17];
  const float* ad2 = (const float*)d_in[18];
  const float* b2  = (const float*)d_in[19];
  const float* ow  = (const float*)d_in[20];
  const float* ob  = (const float*)d_in[21];
  float* y = (float*)d_out;

  const int n = in_sizes[0] / 128;   // 50000
  const int E = in_sizes[1] / 2;     // 400000
  const int Etot = E + n;            // edges + self loops

  float* ws = (float*)d_ws;
  size_t off = 0;
  float* hA    = ws + off; off += (size_t)n * 256;      // pre-agg features
  float* hB    = ws + off; off += (size_t)n * 256;      // aggregated / layer out
  float* esrc  = ws + off; off += (size_t)n * 4;
  float* edst  = ws + off; off += (size_t)n * 4;
  float* mmax  = ws + off; off += (size_t)n * 4;
  float* denom = ws + off; off += (size_t)n * 4;
  float* exbuf = ws + off; off += (size_t)Etot * 4;
  float* colsum = ws + off; off += 256;
  float* shiftb = ws + off; off += 256;
  float* varsum = ws + off; off += 256;
  float* rstdb  = ws + off; off += 256;

  const int eBlocks = (Etot + 255) / 256;
  const float NEG_INF = -__builtin_huge_valf();
  const int rowsPer = 256;
  const int csBlocks = (n + rowsPer - 1) / rowsPer;
  const int mStrips = n / 16;                    // 3125
  const int gemmBx  = (mStrips + 7) / 8;         // 8 waves per block

  // ---------------- Layer 0: IN=128 -> D=256, H=4, C=64 ----------------
  gat_gemm_wmma<<<dim3(gemmBx, 256 / 64), 256, 0, stream>>>(x, W0, hA, 128, 256, n);
  gat_att_scores<<<(n * 4 + 255) / 256, 256, 0, stream>>>(hA, as0, ad0, esrc, edst, n, 4, 64);
  gat_fill<<<(n * 4 + 255) / 256, 256, 0, stream>>>(mmax, NEG_INF, n * 4);
  gat_fill<<<(n * 4 + 255) / 256, 256, 0, stream>>>(denom, 0.f, n * 4);
  gat_fill<<<((size_t)n * 256 + 255) / 256, 256, 0, stream>>>(hB, 0.f, n * 256);
  gat_edge_max<<<eBlocks, 256, 0, stream>>>(ei, E, n, esrc, edst, mmax, 4);
  gat_edge_expsum<<<eBlocks, 256, 0, stream>>>(ei, E, n, esrc, edst, mmax, exbuf, denom, 4);
  gat_edge_agg<<<Etot, 256, 0, stream>>>(ei, E, n, hA, exbuf, denom, hB, 4, 64);
  // GraphNorm + ReLU on hB (bias folded into shift)
  gat_fill<<<1, 256, 0, stream>>>(colsum, 0.f, 256);
  gat_col_accum<<<csBlocks, 256, 0, stream>>>(hB, colsum, n, 256, rowsPer, 0, shiftb);
  gat_prep_shift<<<1, 256, 0, stream>>>(colsum, b0, gm0, shiftb, n);
  gat_fill<<<1, 256, 0, stream>>>(varsum, 0.f, 256);
  gat_col_accum<<<csBlocks, 256, 0, stream>>>(hB, varsum, n, 256, rowsPer, 1, shiftb);
  gat_prep_rstd<<<1, 256, 0, stream>>>(varsum, rstdb, n);
  gat_apply_norm<<<((size_t)n * 256 + 255) / 256, 256, 0, stream>>>(hB, shiftb, rstdb, gw0, gb0, n * 256, 256);

  // ---------------- Layer 1: 256 -> 256, H=4, C=64 ----------------
  gat_gemm_wmma<<<dim3(gemmBx, 256 / 64), 256, 0, stream>>>(hB, W1, hA, 256, 256, n);
  gat_att_scores<<<(n * 4 + 255) / 256, 256, 0, stream>>>(hA, as1, ad1, esrc, edst, n, 4, 64);
  gat_fill<<<(n * 4 + 255) / 256, 256, 0, stream>>>(mmax, NEG_INF, n * 4);
  gat_fill<<<(n * 4 + 255) / 256, 256, 0, stream>>>(denom, 0.f, n * 4);
  gat_fill<<<((size_t)n * 256 + 255) / 256, 256, 0, stream>>>(hB, 0.f, n * 256);
  gat_edge_max<<<eBlocks, 256, 0, stream>>>(ei, E, n, esrc, edst, mmax, 4);
  gat_edge_expsum<<<eBlocks, 256, 0, stream>>>(ei, E, n, esrc, edst, mmax, exbuf, denom, 4);
  gat_edge_agg<<<Etot, 256, 0, stream>>>(ei, E, n, hA, exbuf, denom, hB, 4, 64);
  gat_fill<<<1, 256, 0, stream>>>(colsum, 0.f, 256);
  gat_col_accum<<<csBlocks, 256, 0, stream>>>(hB, colsum, n, 256, rowsPer, 0, shiftb);
  gat_prep_shift<<<1, 256, 0, stream>>>(colsum, b1, gm1, shiftb, n);
  gat_fill<<<1, 256, 0, stream>>>(varsum, 0.f, 256);
  gat_col_accum<<<csBlocks, 256, 0, stream>>>(hB, varsum, n, 256, rowsPer, 1, shiftb);
  gat_prep_rstd<<<1, 256, 0, stream>>>(varsum, rstdb, n);
  gat_apply_norm<<<((size_t)n * 256 + 255) / 256, 256, 0, stream>>>(hB, shiftb, rstdb, gw1, gb1, n * 256, 256);

  // ---------------- Layer 2: 256 -> 128, H=1, C=128, no norm ----------------
  float* h2   = hA;                   // [n,128]
  float* agg2 = hA + (size_t)n * 128; // [n,128]
  gat_gemm_wmma<<<dim3(gemmBx, 128 / 64), 256, 0, stream>>>(hB, W2, h2, 256, 128, n);
  gat_att_scores<<<(n + 255) / 256, 256, 0, stream>>>(h2, as2, ad2, esrc, edst, n, 1, 128);
  gat_fill<<<(n + 255) / 256, 256, 0, stream>>>(mmax, NEG_INF, n);
  gat_fill<<<(n + 255) / 256, 256, 0, stream>>>(denom, 0.f, n);
  gat_fill<<<((size_t)n * 128 + 255) / 256, 256, 0, stream>>>(agg2, 0.f, n * 128);
  gat_edge_max<<<eBlocks, 256, 0, stream>>>(ei, E, n, esrc, edst, mmax, 1);
  gat_edge_expsum<<<eBlocks, 256, 0, stream>>>(ei, E, n, esrc, edst, mmax, exbuf, denom, 1);
  gat_edge_agg<<<Etot, 128, 0, stream>>>(ei, E, n, h2, exbuf, denom, agg2, 1, 128);

  // ---------------- Output linear ----------------
  gat_final_out<<<(n + 255) / 256, 256, 0, stream>>>(agg2, b2, ow, ob, y, n);
}